// ContinuousConvModel_34411277976255
// MI455X (gfx1250) — compile-verified
//
#include <hip/hip_runtime.h>

// ---------------------------------------------------------------------------
// ContinuousConv graph model, MI455X (gfx1250, wave32).
//   K0: zero accumulators in d_ws
//   K1: edge kernel  - filters (64KB) in LDS via global_load_async_to_lds_b128,
//                      wave-per-edge, lanes = output-channel pairs,
//                      f32 atomics scatter into hacc/cnt
//   K2: node kernel  - tanh(mean), concat, LayerNorm, 68x3 projection via
//                      V_WMMA_F32_16X16X4_F32 (17 chained K-steps)
// ---------------------------------------------------------------------------

#define D_GRID   4
#define IN_CH    4
#define CC_DIM   64
#define OUT_CH   3
#define FILT_DW  (D_GRID*D_GRID*D_GRID*IN_CH*CC_DIM)   // 16384 dwords = 64KB
#define EBLK     256
#define NBLK     256

typedef float v2f __attribute__((ext_vector_type(2)));
typedef float v8f __attribute__((ext_vector_type(8)));
typedef int   v4i __attribute__((ext_vector_type(4)));

// async-load builtin takes int4 pointers qualified with explicit addr spaces
typedef __attribute__((address_space(1))) v4i v4i_g;   // global
typedef __attribute__((address_space(3))) v4i v4i_l;   // LDS

__device__ __forceinline__ void atomAddF(float* p, float v) {
#if defined(__HIP_DEVICE_COMPILE__)
  unsafeAtomicAdd(p, v);   // hardware global_atomic_add_f32
#else
  (void)p; (void)v;
#endif
}

__global__ void cc_zero_kernel(float* __restrict__ p, int n) {
  int i = blockIdx.x * blockDim.x + threadIdx.x;
  if (i < n) p[i] = 0.0f;
}

// ---------------------------------------------------------------------------
// Edge kernel: one edge per wave. Filters staged in LDS.
// LDS filter addressing: cell = ((a*4+b)*4+c)*256 ; + i*64 + o.
// Lane l owns outputs o = 2l, 2l+1 -> ds_load_b64, conflict-free.
// ---------------------------------------------------------------------------
__global__ __launch_bounds__(EBLK) void cc_edge_kernel(
    const float* __restrict__ x, const int* __restrict__ ei,
    const float* __restrict__ filters, float* __restrict__ hacc,
    float* __restrict__ cnt, int N, int E)
{
  __shared__ __align__(16) float ldsF[FILT_DW];
  const int tid = threadIdx.x;

  {   // stage 64KB of filters into LDS (4096 x b128)
    const float4* g4 = (const float4*)filters;
#if defined(__gfx1250__) && __has_builtin(__builtin_amdgcn_global_load_async_to_lds_b128) && __has_builtin(__builtin_amdgcn_s_wait_asynccnt)
    for (int i = tid; i < FILT_DW / 4; i += EBLK) {
      __builtin_amdgcn_global_load_async_to_lds_b128(
          (v4i_g*)(g4 + i), (v4i_l*)(ldsF + 4 * i), 0, 0);
    }
    __builtin_amdgcn_s_wait_asynccnt(0);
#else
    for (int i = tid; i < FILT_DW / 4; i += EBLK) {
      float4 v = g4[i];
      *(float4*)(ldsF + 4 * i) = v;
    }
#endif
  }
  __syncthreads();

  const int lane  = tid & 31;
  const int wave  = blockIdx.x * (EBLK / 32) + (tid >> 5);
  const int nwave = gridDim.x * (EBLK / 32);

  for (int e = wave; e < E; e += nwave) {
    const int row = ei[e];
    const int col = ei[E + e];
    const float* xr = x + (size_t)row * 7;
    const float* xc = x + (size_t)col * 7;
    const float rx = xc[0] - xr[0];
    const float ry = xc[1] - xr[1];
    const float rz = xc[2] - xr[2];
    const float d2 = rx * rx + ry * ry + rz * rz;

    if (lane == 0) atomAddF(&cnt[row], 1.0f);   // counts ALL edges (ref semantics)

    if (d2 < 0.25f) {                            // window==0 otherwise -> skip conv
      const float t      = 1.0f - d2 * 4.0f;     // 1 - d2/R^2, R=0.5
      const float window = t * t * t;
      const float nrm    = sqrtf(d2);
      const float s      = tanhf(nrm) / (nrm + 1e-8f);
      // gc = (mapped+1)*1.5 ; a<-z, b<-y, c<-x
      const float ga = (rz * s + 1.0f) * 1.5f;
      const float gb = (ry * s + 1.0f) * 1.5f;
      const float gc = (rx * s + 1.0f) * 1.5f;

      int a0 = min(max((int)floorf(ga), 0), 3);
      int b0 = min(max((int)floorf(gb), 0), 3);
      int c0 = min(max((int)floorf(gc), 0), 3);
      int ia[2] = {a0, min(a0 + 1, 3)};
      int ib[2] = {b0, min(b0 + 1, 3)};
      int ic[2] = {c0, min(c0 + 1, 3)};
      float ad = ga - (float)a0, bd = gb - (float)b0, cd = gc - (float)c0;
      float wa[2] = {1.0f - ad, ad};
      float wb[2] = {1.0f - bd, bd};
      float wc[2] = {1.0f - cd, cd};

      const float f[4] = {xc[0], xc[1], xc[2], xc[6]};   // feat[col]

      float acc0 = 0.0f, acc1 = 0.0f;
      const float* Fl = ldsF + 2 * lane;
#pragma unroll
      for (int j = 0; j < 8; ++j) {
        const int ja = j >> 2, jb = (j >> 1) & 1, jc = j & 1;
        const float w = window * wa[ja] * wb[jb] * wc[jc];
        const float* Fc = Fl + (((ia[ja] * 4 + ib[jb]) * 4 + ic[jc]) << 8);
#pragma unroll
        for (int i = 0; i < IN_CH; ++i) {
          const float coef = w * f[i];
          const float2 F = *(const float2*)(Fc + i * CC_DIM);  // ds_load_b64
          acc0 = fmaf(coef, F.x, acc0);
          acc1 = fmaf(coef, F.y, acc1);
        }
      }
      float* hp = hacc + (size_t)row * CC_DIM + 2 * lane;
      atomAddF(hp,     acc0);
      atomAddF(hp + 1, acc1);
    }
  }
}

// ---------------------------------------------------------------------------
// Node kernel: one wave per 16 nodes. cat[16][68] in LDS (stride 68 ->
// conflict-free). LN with 2 lanes/node + shfl_xor(16). Projection with
// V_WMMA_F32_16X16X4_F32, 17 K-steps (68 = 17*4 exactly).
//
// Layouts used (ISA 7.12.2):
//   A 16x4 f32 : lanes0-15 M=lane {v0=K0,v1=K1}; lanes16-31 M=lane-16 {K2,K3}
//   B  4x16 f32: symmetric split: v0 = rowK0 | rowK2, v1 = rowK1 | rowK3
//   D 16x16 f32: vgpr r -> (M=r, N=lane) lanes0-15 ; (M=r+8) lanes16-31
// ---------------------------------------------------------------------------
__global__ __launch_bounds__(NBLK) void cc_node_kernel(
    const float* __restrict__ x, const float* __restrict__ hacc,
    const float* __restrict__ cnt, const float* __restrict__ gamma,
    const float* __restrict__ beta, const float* __restrict__ Wout,
    const float* __restrict__ bout, float* __restrict__ out, int N)
{
  __shared__ __align__(16) float Wlds[68 * 16];   // W_out zero-padded to 16 cols
  __shared__ float gl[68], bl[68];
  __shared__ float catL[8][16 * 68];              // per-wave cat tiles

  const int tid = threadIdx.x;
  for (int i = tid; i < 68 * 16; i += NBLK) {
    const int k = i >> 4, n = i & 15;
    Wlds[i] = (n < OUT_CH) ? Wout[k * OUT_CH + n] : 0.0f;
  }
  for (int i = tid; i < 68; i += NBLK) { gl[i] = gamma[i]; bl[i] = beta[i]; }
  __syncthreads();

  const int lane = tid & 31;
  const int w    = tid >> 5;
  const int base = (blockIdx.x * 8 + w) * 16;
  float* cw = catL[w];

  // ---- build cat = [enc(4), tanh(hacc/cnt)(64)] for 16 nodes ----
  for (int m = 0; m < 16; ++m) {
    const int node = min(base + m, N - 1);
    const float inv = 1.0f / fmaxf(cnt[node], 1.0f);
    const float* hp = hacc + (size_t)node * CC_DIM;
    cw[m * 68 + 4 + lane]      = tanhf(hp[lane] * inv);
    cw[m * 68 + 4 + lane + 32] = tanhf(hp[lane + 32] * inv);
    if (lane < 4)
      cw[m * 68 + lane] = x[(size_t)node * 7 + ((lane == 3) ? 6 : lane)];
  }
  __builtin_amdgcn_wave_barrier();

  // ---- LayerNorm over 68, lanes (m, m+16) cooperate on node m ----
  {
    const int m = lane & 15, half = lane >> 4;
    float* cm = cw + m * 68 + half * 34;
    float s = 0.0f;
#pragma unroll
    for (int j = 0; j < 34; ++j) s += cm[j];
    s += __shfl_xor(s, 16);
    const float mu = s * (1.0f / 68.0f);
    float v = 0.0f;
#pragma unroll
    for (int j = 0; j < 34; ++j) { const float d = cm[j] - mu; v += d * d; }
    v += __shfl_xor(v, 16);
    const float rs = rsqrtf(v * (1.0f / 68.0f) + 1e-5f);
#pragma unroll
    for (int j = 0; j < 34; ++j) {
      const int ch = half * 34 + j;
      cm[j] = (cm[j] - mu) * rs * gl[ch] + bl[ch];
    }
  }
  __builtin_amdgcn_wave_barrier();

  // ---- projection: D[16 nodes, 16(3 used)] = normed[16,68] @ Wpad[68,16] ----
  {
    const int nn = lane & 15, half = lane >> 4;
    v8f c = {0.f, 0.f, 0.f, 0.f, 0.f, 0.f, 0.f, 0.f};
#pragma unroll
    for (int k0 = 0; k0 < 68; k0 += 4) {
      const int kk = k0 + 2 * half;
      v2f a, b;
      a.x = cw[nn * 68 + kk];          // A: row nn, K=kk
      a.y = cw[nn * 68 + kk + 1];
      b.x = Wlds[kk * 16 + nn];        // B: K=kk, col nn
      b.y = Wlds[(kk + 1) * 16 + nn];
      c = __builtin_amdgcn_wmma_f32_16x16x4_f32(
          false, a, false, b, (short)0, c, false, false);
    }
    if (nn < OUT_CH) {
      const float bb = bout[nn];
#pragma unroll
      for (int r = 0; r < 8; ++r) {
        const int node = base + r + half * 8;       // D vgpr r -> row r (+8 hi)
        if (node < N) out[(size_t)node * OUT_CH + nn] = c[r] + bb;
      }
    }
  }
}

// ---------------------------------------------------------------------------
extern "C" void kernel_launch(void* const* d_in, const int* in_sizes, int n_in,
                              void* d_out, int out_size, void* d_ws, size_t ws_size,
                              hipStream_t stream)
{
  const float* x       = (const float*)d_in[0];
  const int*   ei      = (const int*)  d_in[1];
  const float* filters = (const float*)d_in[2];
  const float* gamma   = (const float*)d_in[3];
  const float* beta    = (const float*)d_in[4];
  const float* Wout    = (const float*)d_in[5];
  const float* bout    = (const float*)d_in[6];

  const int N = in_sizes[0] / 7;
  const int E = in_sizes[1] / 2;

  float* hacc = (float*)d_ws;              // N * 64
  float* cnt  = hacc + (size_t)N * CC_DIM; // N

  const int ztot = N * (CC_DIM + 1);
  cc_zero_kernel<<<(ztot + 255) / 256, 256, 0, stream>>>(hacc, ztot);

  const int eblocks = 1024;              // 8192 waves, ~49 edges/wave
  cc_edge_kernel<<<eblocks, EBLK, 0, stream>>>(x, ei, filters, hacc, cnt, N, E);

  const int nblocks = (N + 127) / 128;   // 8 waves/block * 16 nodes/wave
  cc_node_kernel<<<nblocks, NBLK, 0, stream>>>(x, hacc, cnt, gamma, beta,
                                               Wout, bout, (float*)d_out, N);
}